// CurveEval_1116691497254
// MI455X (gfx1250) — compile-verified
//
#include <hip/hip_runtime.h>

typedef __attribute__((ext_vector_type(2))) float v2f;
typedef __attribute__((ext_vector_type(8))) float v8f;

#define ORDER 3
#define NCTRL 64
#define PADB  68            // padded per-bsub stride in stage (floats)
#define STAGE_PER_WAVE 272  // 4 * PADB, 16B-aligned chunk per wave

// ---------------------------------------------------------------------------
// Kernel 1: per 16-sample tile, compute kbase (stored pre-scaled to bytes,
// kb*16) and a dense 16x8 basis tile. One wave per tile.
// ---------------------------------------------------------------------------
__global__ void nurbs_prep_tiles(const int* __restrict__ span,
                                 const float* __restrict__ basis,
                                 int* __restrict__ wsK,
                                 float* __restrict__ wsA) {
    const int tile = blockIdx.x;
    const int lane = threadIdx.x;      // 0..31
    const int m    = lane & 15;
    const int s    = tile * 16 + m;

    const int sp = span[s];
    int kb = sp - ORDER;
    #pragma unroll
    for (int mask = 16; mask >= 1; mask >>= 1) {
        int o = __shfl_xor(kb, mask, 32);
        kb = (o < kb) ? o : kb;
    }
    if (kb > NCTRL - 8) kb = NCTRL - 8;
    if (kb < 0) kb = 0;

    if (lane == 0) wsK[tile] = kb * 16;   // pre-scaled: bytes into a CP row block

    if (lane < 16) {
        float a[8];
        #pragma unroll
        for (int i = 0; i < 8; ++i) a[i] = 0.0f;
        const int off = sp - ORDER - kb;  // 0..4 by construction
        #pragma unroll
        for (int j = 0; j <= ORDER; ++j) {
            int o = off + j;
            if (o >= 0 && o < 8) a[o] = basis[s * (ORDER + 1) + j];
        }
        float* dst = wsA + tile * 128 + m * 8;
        #pragma unroll
        for (int i = 0; i < 8; ++i) dst[i] = a[i];
    }
}

// ---------------------------------------------------------------------------
// Kernel 2: one block per 4 batches (256 thr = 8 waves). Per s-tile and wave:
//   D[16x16] = A[16x8] x B[8x16] via two v_wmma_f32_16x16x4_f32
//   (M = 16 samples, N = 4 batches x 4 channels, K = 8-wide basis window),
// then a wave-local LDS transpose -> two (x,y,z,w) points per lane,
// v_rcp_f32 + 2 Newton steps, contiguous b96 stores via per-lane pointers
// bumped by constants. Scalar loop counter; EXEC all-1s throughout.
// ---------------------------------------------------------------------------
__global__ void __launch_bounds__(256)
nurbs_eval_wmma(const float* __restrict__ cp,
                const int* __restrict__ wsK,
                const float* __restrict__ wsA,
                float* __restrict__ out,
                int S) {
    __shared__ __align__(16) float ldscp[4 * NCTRL * 4];       // 4 KB
    __shared__ __align__(16) float stage[8 * STAGE_PER_WAVE];  // 8.5 KB

    const int b_base = blockIdx.x * 4;

    {   // stage control points: 256 float4s, one per thread
        const float4* g4 = (const float4*)cp + (size_t)b_base * NCTRL;
        ((float4*)ldscp)[threadIdx.x] = g4[threadIdx.x];
    }
    __syncthreads();

    const int lane = threadIdx.x & 31;
    const int wid  = threadIdx.x >> 5;

    const int m    = lane & 15;   // A row / B-C-D column
    const int half = lane >> 4;   // K-pair select / row-half select
    const int c    = m & 3;       // channel of this column
    const int bsub = m >> 2;      // batch subgroup of this column

    // LDS byte base for B gathers: add kb16 (runtime, uniform per wave)
    const char* bbase = (const char*)ldscp
                      + (unsigned)(bsub * (NCTRL * 4) + c) * 4u
                      + (unsigned)half * 32u;

    // per-wave staging buffer
    float* st = stage + wid * STAGE_PER_WAVE;
    const int stw = bsub * PADB + half * 32 + c;          // store idx (+4 per row)
    const float* r0 = &st[(lane >> 4) * PADB + m * 4];    // point (pb0, row=m)

    // streaming pointers / offsets with constant per-iteration increments
    const float* ap = wsA + wid * 128 + m * 8 + half * 2;
    int koff = wid;
    float* op0 = out + ((size_t)(b_base + (lane >> 4)) * S + m) * 3 + wid * 48;
    float* op1 = op0 + 2 * (size_t)S * 3;

    const int iters = S >> 7;   // (S/16 tiles) / 8 waves

    for (int it = 0; it < iters; ++it) {
        const int kb16 = wsK[koff];
        const float* bp = (const float*)(bbase + (unsigned)kb16);

        v2f av0 = { ap[0], ap[1] };
        v2f av1 = { ap[4], ap[5] };
        v2f bv0 = { bp[0],  bp[4]  };
        v2f bv1 = { bp[16], bp[20] };

        v8f acc = {0.f, 0.f, 0.f, 0.f, 0.f, 0.f, 0.f, 0.f};
        acc = __builtin_amdgcn_wmma_f32_16x16x4_f32(false, av0, false, bv0,
                                                    (short)0, acc, false, false);
        acc = __builtin_amdgcn_wmma_f32_16x16x4_f32(false, av1, false, bv1,
                                                    (short)0, acc, false, false);

        // wave-local transpose through LDS (bank-conflict-free, wave-in-order)
        #pragma unroll
        for (int i = 0; i < 8; ++i)
            st[stw + i * 4] = acc[i];

        __builtin_amdgcn_wave_barrier();

        const float4 v0 = *(const float4*)r0;
        const float4 v1 = *(const float4*)(r0 + 2 * PADB);

        float rc0 = __builtin_amdgcn_rcpf(v0.w);
        rc0 = __builtin_fmaf(__builtin_fmaf(-v0.w, rc0, 1.0f), rc0, rc0);
        rc0 = __builtin_fmaf(__builtin_fmaf(-v0.w, rc0, 1.0f), rc0, rc0);
        float rc1 = __builtin_amdgcn_rcpf(v1.w);
        rc1 = __builtin_fmaf(__builtin_fmaf(-v1.w, rc1, 1.0f), rc1, rc1);
        rc1 = __builtin_fmaf(__builtin_fmaf(-v1.w, rc1, 1.0f), rc1, rc1);

        op0[0] = v0.x * rc0;
        op0[1] = v0.y * rc0;
        op0[2] = v0.z * rc0;
        op1[0] = v1.x * rc1;
        op1[1] = v1.y * rc1;
        op1[2] = v1.z * rc1;

        __builtin_amdgcn_wave_barrier();

        ap   += 1024;    // 8 tiles * 128 floats
        koff += 8;
        op0  += 384;     // 8 tiles * 16 samples * 3 floats
        op1  += 384;
    }
}

extern "C" void kernel_launch(void* const* d_in, const int* in_sizes, int n_in,
                              void* d_out, int out_size, void* d_ws, size_t ws_size,
                              hipStream_t stream) {
    const float* cp    = (const float*)d_in[0];   // [B, 64, 4] f32
    const int*   span  = (const int*)d_in[1];     // [S] i32
    const float* basis = (const float*)d_in[2];   // [S, 4] f32

    const int B = in_sizes[0] / (NCTRL * 4);
    const int S = in_sizes[1];
    const int ntiles = S / 16;

    int*   wsK = (int*)d_ws;
    size_t aoff = ((size_t)ntiles * sizeof(int) + 255) & ~(size_t)255;
    float* wsA = (float*)((char*)d_ws + aoff);

    nurbs_prep_tiles<<<ntiles, 32, 0, stream>>>(span, basis, wsK, wsA);
    nurbs_eval_wmma<<<B / 4, 256, 0, stream>>>(cp, wsK, wsA, (float*)d_out, S);
}